// PointMetaBaseEncoder_65910568124555
// MI455X (gfx1250) — compile-verified
//
#include <hip/hip_runtime.h>
#include <hip/hip_bf16.h>
#include <cstdint>

namespace {

constexpr int B_ = 4, N_ = 8192, CIN_ = 4, C_ = 64, K_ = 32, STRIDE_ = 4, NB_ = 2, EXP_ = 4;
constexpr int M_ = N_ / STRIDE_;
constexpr float R1SQ = 0.1f * 0.1f;
constexpr float R2SQ = 0.2f * 0.2f;
constexpr float NEG_BIG = -3.4028235e38f;

typedef float v2f __attribute__((ext_vector_type(2)));
typedef float v8f __attribute__((ext_vector_type(8)));

// ---------------------------------------------------------------------------
// f1 = relu(f^T @ W_sa + b_sa) as a K=4 WMMA: one V_WMMA_F32_16X16X4_F32 per
// 16x16 tile. Rows are (b*N + n); N % 16 == 0 so a tile never crosses b.
//   A(16x4): lane = khi*16 + m, vgpr0/1 = f[b, 2*khi + {0,1}, n0+m]
//   B(4x16): lane = khi*16 + n, vgpr0/1 = W_sa[2*khi + {0,1}, c0+n]
// ---------------------------------------------------------------------------
__global__ void f1_wmma_kernel(const float* __restrict__ f, const float* __restrict__ Wsa,
                               const float* __restrict__ bsa, float* __restrict__ f1) {
  int wave = (blockIdx.x * blockDim.x + threadIdx.x) >> 5;
  int lane = threadIdx.x & 31;
  constexpr int tilesC = C_ / 16;                       // 4
  constexpr int nTiles = (B_ * N_ / 16) * tilesC;       // 8192
  if (wave >= nTiles) return;                           // wave-uniform
  int row0 = (wave / tilesC) * 16;                      // global row (b*N + n)
  int c0 = (wave % tilesC) * 16;
  int b = row0 / N_;
  int n0 = row0 % N_;
  int half = lane >> 4;
  int ln = lane & 15;

  const float* fb = f + ((size_t)b * CIN_ + 2 * half) * N_ + n0 + ln;
  v2f a;  a.x = fb[0];        a.y = fb[N_];             // ci = 2*half, 2*half+1
  const float* wc = Wsa + (size_t)(2 * half) * C_ + c0 + ln;
  v2f bb; bb.x = wc[0];       bb.y = wc[C_];
  v8f acc = {0.f, 0.f, 0.f, 0.f, 0.f, 0.f, 0.f, 0.f};
  acc = __builtin_amdgcn_wmma_f32_16x16x4_f32(false, a, false, bb, (short)0, acc,
                                              false, false);
  int col = c0 + ln;
  float bv = bsa[col];
#pragma unroll
  for (int r = 0; r < 8; ++r) {
    int row = row0 + r + 8 * half;
    f1[(size_t)row * C_ + col] = fmaxf(acc[r] + bv, 0.0f);
  }
}

// ---------------------------------------------------------------------------
// Ball query: one wave32 per query q[b,m] = p[b, m*STRIDE]; source j at
// p[b, j*src_stride]. First K ascending indices with d^2 <= r2, padded with
// the first hit (a hit always exists: the query is in the source set).
// ---------------------------------------------------------------------------
__global__ void ball_query_kernel(const float* __restrict__ p, int n_src, int src_stride,
                                  float r2, int* __restrict__ idx_out) {
  int wave = (blockIdx.x * blockDim.x + threadIdx.x) >> 5;
  int lane = threadIdx.x & 31;
  if (wave >= B_ * M_) return;
  int b = wave / M_;
  int m = wave % M_;
  const float* pb = p + (size_t)b * N_ * 3;
  float qx = pb[(size_t)m * STRIDE_ * 3 + 0];
  float qy = pb[(size_t)m * STRIDE_ * 3 + 1];
  float qz = pb[(size_t)m * STRIDE_ * 3 + 2];
  int* out = idx_out + (size_t)wave * K_;

  int count = 0;
  int first_idx = n_src - 1;
  for (int base = 0; base < n_src; base += 32) {        // n_src % 32 == 0
    int j = base + lane;
    const float* sp = pb + (size_t)j * src_stride * 3;
    float dx = sp[0] - qx, dy = sp[1] - qy, dz = sp[2] - qz;
    bool hit = (dx * dx + dy * dy + dz * dz) <= r2;
    unsigned mask = (unsigned)__ballot(hit);            // wave32 -> low 32 bits
    if (count == 0 && mask != 0u) first_idx = base + (__ffs(mask) - 1);
    if (hit) {
      int pos = count + __popc(mask & ((1u << lane) - 1u));
      if (pos < K_) out[pos] = j;
    }
    count += __popc(mask);
    if (count >= K_) break;
  }
  if (count > K_) count = K_;
  for (int t = count + lane; t < K_; t += 32) out[t] = first_idx;
}

// ---------------------------------------------------------------------------
// Neighborhood max with positional encoding, LDS-staged geometry.
// Block = 256 threads = 4 query groups x 64 channels. Per group, 32 threads
// stage neighbor index + (dx,dy,dz) into LDS once; then all 64 channel
// threads read LDS (ds_load) and do coalesced gathers of src[j*C + c].
//   out[bm*C + c] = max_k( dq . Wpe[:,c] + bpe[c] + src[(boff + j)*C + c] )
// src_stride: 1 -> neighbors are p-points (feat stage), STRIDE_ -> q-points.
// ---------------------------------------------------------------------------
__global__ void neigh_max_kernel(const float* __restrict__ p, const float* __restrict__ src,
                                 const int* __restrict__ idx, int src_stride, int src_n,
                                 const float* __restrict__ Wpe, const float* __restrict__ bpe,
                                 float* __restrict__ out) {
  __shared__ int s_idx[4][K_];
  __shared__ float s_dq[4][K_][3];
  int g = threadIdx.x >> 6;                             // query group in block
  int c = threadIdx.x & 63;                             // channel
  int bm = blockIdx.x * 4 + g;
  if (bm >= B_ * M_) return;
  int b = bm / M_;
  int m = bm % M_;
  const float* pb = p + (size_t)b * N_ * 3;

  if (c < K_) {
    float qx = pb[(size_t)m * STRIDE_ * 3 + 0];
    float qy = pb[(size_t)m * STRIDE_ * 3 + 1];
    float qz = pb[(size_t)m * STRIDE_ * 3 + 2];
    int j = idx[(size_t)bm * K_ + c];
    s_idx[g][c] = j;
    const float* sp = pb + (size_t)j * src_stride * 3;
    s_dq[g][c][0] = sp[0] - qx;
    s_dq[g][c][1] = sp[1] - qy;
    s_dq[g][c][2] = sp[2] - qz;
  }
  __syncthreads();

  float w0 = Wpe[0 * C_ + c], w1 = Wpe[1 * C_ + c], w2 = Wpe[2 * C_ + c];
  float bias = bpe[c];
  const float* sb = src + (size_t)b * src_n * C_;
  float best = NEG_BIG;
#pragma unroll 4
  for (int k = 0; k < K_; ++k) {
    int j = s_idx[g][k];
    float v = s_dq[g][k][0] * w0 + s_dq[g][k][1] * w1 + s_dq[g][k][2] * w2 + bias
            + sb[(size_t)j * C_ + c];
    best = fmaxf(best, v);
  }
  out[(size_t)bm * C_ + c] = best;
}

// ---------------------------------------------------------------------------
// Register-blocked fp32 WMMA GEMM:
//   out(Mrows x Ncols) = act( X(Mrows x KDIM) @ W(KDIM x Ncols) + bias [+skip] )
// One wave owns a 64x16 output tile (4 M sub-tiles). Per K-step of 4, the B
// operand is loaded once and reused by 4 V_WMMA_F32_16X16X4_F32 ops.
//   A(16x4): lane = khi*16 + m, vgpr0/1 = X[m][k + 2*khi + {0,1}]
//   B(4x16): lane = khi*16 + n, vgpr0/1 = W[k + 2*khi + {0,1}][n]
//   C/D:     vgpr r, lane = khi*16 + n -> row = r + 8*khi, col = n
// ---------------------------------------------------------------------------
template <int KDIM, bool RELU, bool SKIP>
__global__ void gemm_wmma_kernel(const float* __restrict__ X, const float* __restrict__ W,
                                 const float* __restrict__ bias,
                                 const float* __restrict__ skip, float* __restrict__ out,
                                 int Mrows, int Ncols) {
  constexpr int MT = 4;                                 // M sub-tiles per wave
  int wave = (blockIdx.x * blockDim.x + threadIdx.x) >> 5;
  int lane = threadIdx.x & 31;
  int tilesN = Ncols >> 4;
  int nTiles = (Mrows >> 6) * tilesN;                   // 64-row macro tiles
  if (wave >= nTiles) return;                           // wave-uniform
  int m0 = (wave / tilesN) << 6;
  int n0 = (wave % tilesN) << 4;
  int half = lane >> 4;
  int ln = lane & 15;

  const float* Xrow0 = X + (size_t)(m0 + ln) * KDIM + 2 * half;
  const float* Wcol = W + (size_t)(2 * half) * Ncols + n0 + ln;
  constexpr size_t XSTEP = (size_t)16 * KDIM;           // 16 rows down

  v8f acc[MT];
#pragma unroll
  for (int t = 0; t < MT; ++t) acc[t] = v8f{0.f, 0.f, 0.f, 0.f, 0.f, 0.f, 0.f, 0.f};

#pragma unroll 4
  for (int k = 0; k < KDIM; k += 4) {
    v2f bb;                                             // loaded once, used 4x
    bb.x = Wcol[(size_t)k * Ncols];
    bb.y = Wcol[(size_t)(k + 1) * Ncols];
#pragma unroll
    for (int t = 0; t < MT; ++t) {
      v2f a;
      a.x = Xrow0[t * XSTEP + k];
      a.y = Xrow0[t * XSTEP + k + 1];
      acc[t] = __builtin_amdgcn_wmma_f32_16x16x4_f32(false, a, false, bb, (short)0,
                                                     acc[t], false, false);
    }
  }

  int col = n0 + ln;
  float bv = bias[col];
#pragma unroll
  for (int t = 0; t < MT; ++t) {
#pragma unroll
    for (int r = 0; r < 8; ++r) {
      int row = m0 + t * 16 + r + 8 * half;
      float v = acc[t][r] + bv;
      if (SKIP) v += skip[(size_t)row * Ncols + col];
      if (RELU) v = fmaxf(v, 0.0f);
      out[(size_t)row * Ncols + col] = v;
    }
  }
}

}  // namespace

// ---------------------------------------------------------------------------
extern "C" void kernel_launch(void* const* d_in, const int* in_sizes, int n_in,
                              void* d_out, int out_size, void* d_ws, size_t ws_size,
                              hipStream_t stream) {
  (void)in_sizes; (void)n_in; (void)out_size; (void)ws_size;
  const float* p      = (const float*)d_in[0];
  const float* f      = (const float*)d_in[1];
  const float* W_sa   = (const float*)d_in[2];
  const float* b_sa   = (const float*)d_in[3];
  const float* W_pe1  = (const float*)d_in[4];
  const float* b_pe1  = (const float*)d_in[5];
  const float* W_pe2  = (const float*)d_in[6];
  const float* b_pe2  = (const float*)d_in[7];
  const float* W_conv = (const float*)d_in[8];
  const float* b_conv = (const float*)d_in[9];
  const float* W_pw1  = (const float*)d_in[10];
  const float* b_pw1  = (const float*)d_in[11];
  const float* W_pw2  = (const float*)d_in[12];
  const float* b_pw2  = (const float*)d_in[13];
  float* feat = (float*)d_out;                          // (B, M, C) fp32

  // Workspace carve-up (~23 MB, all 4-byte aligned)
  char* ws = (char*)d_ws;
  float* f1 = (float*)ws;  ws += sizeof(float) * (size_t)B_ * N_ * C_;          // 8 MB
  float* h  = (float*)ws;  ws += sizeof(float) * (size_t)B_ * M_ * C_;          // 2 MB
  float* h2 = (float*)ws;  ws += sizeof(float) * (size_t)B_ * M_ * EXP_ * C_;   // 8 MB
  float* agg = (float*)ws; ws += sizeof(float) * (size_t)B_ * M_ * C_;          // 2 MB
  int* idx1 = (int*)ws;    ws += sizeof(int) * (size_t)B_ * M_ * K_;            // 1 MB
  int* idx2 = (int*)ws;                                                          // 1 MB

  const int THREADS = 256;

  // 1. Input embedding (one WMMA per 16x16 tile, K=4)
  {
    int waves = (B_ * N_ / 16) * (C_ / 16);
    f1_wmma_kernel<<<(waves * 32 + THREADS - 1) / THREADS, THREADS, 0, stream>>>(
        f, W_sa, b_sa, f1);
  }

  // 2. Ball queries (one wave32 per query)
  int bq_blocks = (B_ * M_ * 32 + THREADS - 1) / THREADS;
  ball_query_kernel<<<bq_blocks, THREADS, 0, stream>>>(p, N_, 1, R1SQ, idx1);
  ball_query_kernel<<<bq_blocks, THREADS, 0, stream>>>(p, M_, STRIDE_, R2SQ, idx2);

  // 3. Initial aggregation: feat = max_k(pe1 + f1[idx1])  (neighbors in p)
  int nm_blocks = (B_ * M_ + 3) / 4;
  neigh_max_kernel<<<nm_blocks, THREADS, 0, stream>>>(p, f1, idx1, 1, N_, W_pe1, b_pe1, feat);

  // 4. NB residual blocks
  const int MR = B_ * M_;
  for (int i = 0; i < NB_; ++i) {
    {  // h = relu(feat @ W_conv[i] + b_conv[i])
      int waves = (MR / 64) * (C_ / 16);
      gemm_wmma_kernel<C_, true, false>
          <<<(waves * 32 + THREADS - 1) / THREADS, THREADS, 0, stream>>>(
              feat, W_conv + (size_t)i * C_ * C_, b_conv + (size_t)i * C_, nullptr, h, MR, C_);
    }
    // agg = max_k(pe2 + h[idx2])  (neighbors in q = strided p)
    neigh_max_kernel<<<nm_blocks, THREADS, 0, stream>>>(p, h, idx2, STRIDE_, M_, W_pe2,
                                                        b_pe2, agg);
    {  // h2 = relu(agg @ W_pw1[i] + b_pw1[i])
      int waves = (MR / 64) * ((EXP_ * C_) / 16);
      gemm_wmma_kernel<C_, true, false>
          <<<(waves * 32 + THREADS - 1) / THREADS, THREADS, 0, stream>>>(
              agg, W_pw1 + (size_t)i * C_ * EXP_ * C_, b_pw1 + (size_t)i * EXP_ * C_, nullptr,
              h2, MR, EXP_ * C_);
    }
    {  // feat = relu(h2 @ W_pw2[i] + b_pw2[i] + feat)  (in-place: same thread R/W)
      int waves = (MR / 64) * (C_ / 16);
      gemm_wmma_kernel<EXP_ * C_, true, true>
          <<<(waves * 32 + THREADS - 1) / THREADS, THREADS, 0, stream>>>(
              h2, W_pw2 + (size_t)i * EXP_ * C_ * C_, b_pw2 + (size_t)i * C_, feat, feat, MR, C_);
    }
  }
}